// HeteroGNN_Diagnosis_48971217109307
// MI455X (gfx1250) — compile-verified
//
#include <hip/hip_runtime.h>
#include <hip/hip_bf16.h>
#include <math.h>

// ---------------- problem constants (from reference) ----------------
#define NP_  100000
#define NL_  10000
#define ND_  500
#define E1_  2000000
#define E2_  500000
#define H_   128
#define HH_  (H_ * H_)
#define HOUT_ 10
#define FC1_ 64
#define EPS_ 1e-5f

typedef __attribute__((ext_vector_type(2))) float v2f;
typedef __attribute__((ext_vector_type(8))) float v8f;

// ---------------- utility kernels ----------------
__global__ void k_zero(float* __restrict__ p, long long n) {
    long long i = (long long)blockIdx.x * blockDim.x + threadIdx.x;
    if (i < n) p[i] = 0.0f;
}

// out[N,H] = x[N,K] @ W[K,H] + b[H]   (small K: 1/2/16)
__global__ void k_proj(const float* __restrict__ x, const float* __restrict__ W,
                       const float* __restrict__ b, float* __restrict__ out,
                       int N, int K) {
    long long i = (long long)blockIdx.x * blockDim.x + threadIdx.x;
    if (i >= (long long)N * H_) return;
    int row = (int)(i >> 7);      // /128
    int col = (int)(i & (H_ - 1));
    float acc = b[col];
    for (int k = 0; k < K; ++k)
        acc += x[(long long)row * K + k] * W[(long long)k * H_ + col];
    out[i] = acc;
}

// cnt[dst[e]] += 1
__global__ void k_count(const int* __restrict__ dst, int E, float* __restrict__ cnt) {
    long long e = (long long)blockIdx.x * blockDim.x + threadIdx.x;
    if (e < E) atomicAdd(&cnt[dst[e]], 1.0f);
}

// agg[dst[e], :] += xsrc[src[e], :]   (one wave per edge; lane covers float4)
__global__ void k_scatter(const float* __restrict__ xsrc,
                          const int* __restrict__ src, const int* __restrict__ dst,
                          int E, float* __restrict__ agg) {
    long long tid = (long long)blockIdx.x * blockDim.x + threadIdx.x;
    long long e = tid >> 5;
    if (e >= E) return;
    int lane = (int)(tid & 31);
    int s = src[e], d = dst[e];
    const float4 v = *(const float4*)(xsrc + (long long)s * H_ + lane * 4);
    float* base = agg + (long long)d * H_ + lane * 4;
    atomicAdd(base + 0, v.x);
    atomicAdd(base + 1, v.y);
    atomicAdd(base + 2, v.z);
    atomicAdd(base + 3, v.w);
}

// agg[i] /= max(cnt[i/H], 1)
__global__ void k_mean(float* __restrict__ agg, const float* __restrict__ cnt, int N) {
    long long i = (long long)blockIdx.x * blockDim.x + threadIdx.x;
    if (i >= (long long)N * H_) return;
    float c = cnt[i >> 7];
    agg[i] *= 1.0f / fmaxf(c, 1.0f);
}

// ---------------- fp32 WMMA GEMM: out[N,Hout] = sum_t A_t[N,128] @ W_t[128,Hout] (+biases) ----------------
// 256 threads = 8 waves; wave w computes column tile w (16 cols). A tile staged in LDS per term.
__global__ __launch_bounds__(256)
void k_gemm4(const float* __restrict__ A0, const float* __restrict__ W0,
             const float* __restrict__ A1, const float* __restrict__ W1,
             const float* __restrict__ A2, const float* __restrict__ W2,
             const float* __restrict__ A3, const float* __restrict__ W3,
             const float* __restrict__ bias0, const float* __restrict__ bias1,
             float* __restrict__ out, int N, int Hout, int do_relu) {
    __shared__ float sA[16 * H_];   // 8 KB: one 16-row A stripe
    const int tid  = threadIdx.x;
    const int wave = tid >> 5;
    const int lane = tid & 31;
    const int m    = lane & 15;     // row (A) / col (B) within tile
    const int half = lane >> 4;     // K-half selector
    const int tileM = blockIdx.x * 16;
    const int nCol  = Hout >> 4;

    v8f acc = {};
    const float* As[4] = {A0, A1, A2, A3};
    const float* Ws[4] = {W0, W1, W2, W3};

#pragma unroll
    for (int t = 0; t < 4; ++t) {
        const float* A = As[t];
        if (!A) break;                      // terms are left-packed; uniform
        __syncthreads();
        // cooperatively stage A[tileM:tileM+16, 0:128] into LDS (zero-pad OOB rows)
        for (int i = tid; i < (16 * H_) / 4; i += 256) {
            int r  = i >> 5;                // 32 float4 per row
            int c4 = (i & 31) << 2;
            float4 v = make_float4(0.f, 0.f, 0.f, 0.f);
            int row = tileM + r;
            if (row < N) v = *(const float4*)(A + (long long)row * H_ + c4);
            *(float4*)(&sA[r * H_ + c4]) = v;
        }
        __syncthreads();
        if (wave < nCol) {
            const float* W = Ws[t];
            const int colBase = wave << 4;
            for (int kb = 0; kb < H_; kb += 4) {
                const int k0 = kb + half * 2;
                v2f a, b;
                a.x = sA[m * H_ + k0];
                a.y = sA[m * H_ + k0 + 1];
                b.x = W[(long long)k0 * Hout + colBase + m];
                b.y = W[(long long)(k0 + 1) * Hout + colBase + m];
                acc = __builtin_amdgcn_wmma_f32_16x16x4_f32(
                    false, a, false, b, (short)0, acc, false, false);
            }
        }
    }

    if (wave < nCol) {
        const int colBase = wave << 4;
        float bsum = 0.0f;
        if (bias0) bsum += bias0[colBase + m];
        if (bias1) bsum += bias1[colBase + m];
#pragma unroll
        for (int r = 0; r < 8; ++r) {
            int row = tileM + half * 8 + r;
            if (row < N) {
                float v = acc[r] + bsum;
                if (do_relu) v = fmaxf(v, 0.0f);
                out[(long long)row * Hout + colBase + m] = v;
            }
        }
    }
}

// ---------------- BatchNorm (training mode) ----------------
__global__ void k_bn_stats(const float* __restrict__ x, int N,
                           float* __restrict__ sum, float* __restrict__ sq) {
    int c = threadIdx.x;                 // 128 threads = 128 columns
    float s = 0.f, q = 0.f;
    for (int r = blockIdx.x; r < N; r += gridDim.x) {
        float v = x[(long long)r * H_ + c];
        s += v; q += v * v;
    }
    atomicAdd(&sum[c], s);
    atomicAdd(&sq[c],  q);
}

__global__ void k_bn_prep(const float* __restrict__ sum, const float* __restrict__ sq,
                          const float* __restrict__ g, const float* __restrict__ b,
                          float invN, float* __restrict__ scale, float* __restrict__ shift) {
    int c = threadIdx.x;
    float mean = sum[c] * invN;
    float var  = sq[c] * invN - mean * mean;
    float sc   = g[c] * rsqrtf(var + EPS_);
    scale[c] = sc;
    shift[c] = b[c] - mean * sc;
}

__global__ void k_bn_apply_relu(const float* __restrict__ x,
                                const float* __restrict__ scale,
                                const float* __restrict__ shift,
                                float* __restrict__ y, long long n) {
    long long i = (long long)blockIdx.x * blockDim.x + threadIdx.x;
    if (i >= n) return;
    int c = (int)(i & (H_ - 1));
    y[i] = fmaxf(x[i] * scale[c] + shift[c], 0.0f);
}

// ---------------- head: fc2 + log_softmax ----------------
__global__ void k_head(const float* __restrict__ a /*[NP,64] relu'd*/,
                       const float* __restrict__ W /*[64,10]*/,
                       const float* __restrict__ b /*[10]*/,
                       float* __restrict__ out /*[NP,10]*/) {
    long long r = (long long)blockIdx.x * blockDim.x + threadIdx.x;
    if (r >= NP_) return;
    float logits[HOUT_];
#pragma unroll
    for (int j = 0; j < HOUT_; ++j) logits[j] = b[j];
    for (int k = 0; k < FC1_; ++k) {
        float av = a[r * FC1_ + k];
#pragma unroll
        for (int j = 0; j < HOUT_; ++j)
            logits[j] += av * W[k * HOUT_ + j];
    }
    float mx = logits[0];
#pragma unroll
    for (int j = 1; j < HOUT_; ++j) mx = fmaxf(mx, logits[j]);
    float se = 0.f;
#pragma unroll
    for (int j = 0; j < HOUT_; ++j) se += expf(logits[j] - mx);
    float lse = mx + logf(se);
#pragma unroll
    for (int j = 0; j < HOUT_; ++j) out[r * HOUT_ + j] = logits[j] - lse;
}

// ---------------- host orchestration ----------------
static inline int gridFor(long long n, int b) { return (int)((n + b - 1) / b); }

extern "C" void kernel_launch(void* const* d_in, const int* in_sizes, int n_in,
                              void* d_out, int out_size, void* d_ws, size_t ws_size,
                              hipStream_t stream) {
    (void)in_sizes; (void)n_in; (void)out_size; (void)ws_size;

    const float* x_patient = (const float*)d_in[0];
    const float* x_lab     = (const float*)d_in[1];
    const float* x_disease = (const float*)d_in[2];
    const float* Wp   = (const float*)d_in[3];
    const float* bp   = (const float*)d_in[4];
    const float* Wlab = (const float*)d_in[5];
    const float* blab = (const float*)d_in[6];
    const float* Wdis = (const float*)d_in[7];
    const float* bdis = (const float*)d_in[8];
    const float* Wl1  = (const float*)d_in[9];
    const float* bl1  = (const float*)d_in[10];
    const float* Wr1  = (const float*)d_in[11];
    const float* Wl2  = (const float*)d_in[12];
    const float* bl2  = (const float*)d_in[13];
    const float* Wr2  = (const float*)d_in[14];
    const float* bn1_g = (const float*)d_in[15];
    const float* bn1_b = (const float*)d_in[16];
    const float* bn2_g = (const float*)d_in[17];
    const float* bn2_b = (const float*)d_in[18];
    const float* fc1_W = (const float*)d_in[19];
    const float* fc1_b = (const float*)d_in[20];
    const float* fc2_W = (const float*)d_in[21];
    const float* fc2_b = (const float*)d_in[22];
    const int* e1_src = (const int*)d_in[23];
    const int* e1_dst = (const int*)d_in[24];
    const int* e2_src = (const int*)d_in[25];
    const int* e2_dst = (const int*)d_in[26];

    // ---- workspace carve (floats) ----
    float* ws = (float*)d_ws;
    long long o = 0;
    float* hp    = ws + o; o += (long long)NP_ * H_;
    float* hp2   = ws + o; o += (long long)NP_ * H_;
    float* aggP1 = ws + o; o += (long long)NP_ * H_;
    float* aggP2 = ws + o; o += (long long)NP_ * H_;
    float* hl    = ws + o; o += (long long)NL_ * H_;
    float* hl2   = ws + o; o += (long long)NL_ * H_;
    float* aggL  = ws + o; o += (long long)NL_ * H_;
    float* hd    = ws + o; o += (long long)ND_ * H_;
    float* hd2   = ws + o; o += (long long)ND_ * H_;
    float* aggD  = ws + o; o += (long long)ND_ * H_;
    float* cntL  = ws + o; o += NL_;
    float* cntD  = ws + o; o += ND_;
    float* cntP1 = ws + o; o += NP_;
    float* cntP2 = ws + o; o += NP_;
    float* bn_sum   = ws + o; o += H_;
    float* bn_sq    = ws + o; o += H_;
    float* bn_scale = ws + o; o += H_;
    float* bn_shift = ws + o; o += H_;
    float* fc1out = aggP1;   // reuse [NP,64] <= [NP,128]

    const int B = 256;
    const long long nPH = (long long)NP_ * H_;
    const long long nLH = (long long)NL_ * H_;
    const long long nDH = (long long)ND_ * H_;

    // ---- degree counts (shared by both layers) ----
    k_zero<<<gridFor(NL_, B), B, 0, stream>>>(cntL, NL_);
    k_zero<<<gridFor(ND_, B), B, 0, stream>>>(cntD, ND_);
    k_zero<<<gridFor(NP_, B), B, 0, stream>>>(cntP1, NP_);
    k_zero<<<gridFor(NP_, B), B, 0, stream>>>(cntP2, NP_);
    k_count<<<gridFor(E1_, B), B, 0, stream>>>(e1_dst, E1_, cntL);
    k_count<<<gridFor(E2_, B), B, 0, stream>>>(e2_dst, E2_, cntD);
    k_count<<<gridFor(E1_, B), B, 0, stream>>>(e1_src, E1_, cntP1);
    k_count<<<gridFor(E2_, B), B, 0, stream>>>(e2_src, E2_, cntP2);

    // ---- input projections ----
    k_proj<<<gridFor(nPH, B), B, 0, stream>>>(x_patient, Wp, bp, hp, NP_, 16);
    k_proj<<<gridFor(nLH, B), B, 0, stream>>>(x_lab, Wlab, blab, hl, NL_, 1);
    k_proj<<<gridFor(nDH, B), B, 0, stream>>>(x_disease, Wdis, bdis, hd, ND_, 2);

    // ---- two hetero layers ----
    for (int layer = 0; layer < 2; ++layer) {
        const float* Wl = (layer == 0) ? Wl1 : Wl2;
        const float* bl = (layer == 0) ? bl1 : bl2;
        const float* Wr = (layer == 0) ? Wr1 : Wr2;
        const float* g  = (layer == 0) ? bn1_g : bn2_g;
        const float* bb = (layer == 0) ? bn1_b : bn2_b;

        // labs: out_l = mean(hp over e1)@Wl[0] + bl[0] + hl@Wr[0]
        k_zero<<<gridFor(nLH, B), B, 0, stream>>>(aggL, nLH);
        k_scatter<<<gridFor((long long)E1_ * 32, B), B, 0, stream>>>(hp, e1_src, e1_dst, E1_, aggL);
        k_mean<<<gridFor(nLH, B), B, 0, stream>>>(aggL, cntL, NL_);
        k_gemm4<<<gridFor(NL_, 16), 256, 0, stream>>>(
            aggL, Wl + 0 * HH_, hl, Wr + 0 * HH_, nullptr, nullptr, nullptr, nullptr,
            bl + 0 * H_, nullptr, hl2, NL_, H_, 0);

        // diseases: out_d = mean(hp over e2)@Wl[1] + bl[1] + hd@Wr[1]
        k_zero<<<gridFor(nDH, B), B, 0, stream>>>(aggD, nDH);
        k_scatter<<<gridFor((long long)E2_ * 32, B), B, 0, stream>>>(hp, e2_src, e2_dst, E2_, aggD);
        k_mean<<<gridFor(nDH, B), B, 0, stream>>>(aggD, cntD, ND_);
        k_gemm4<<<gridFor(ND_, 16), 256, 0, stream>>>(
            aggD, Wl + 1 * HH_, hd, Wr + 1 * HH_, nullptr, nullptr, nullptr, nullptr,
            bl + 1 * H_, nullptr, hd2, ND_, H_, 0);

        // patients: out_p = mean(hl rev e1)@Wl[2] + mean(hd rev e2)@Wl[3]
        //                 + hp@Wr[2] + hp@Wr[3] + bl[2] + bl[3]
        k_zero<<<gridFor(nPH, B), B, 0, stream>>>(aggP1, nPH);
        k_scatter<<<gridFor((long long)E1_ * 32, B), B, 0, stream>>>(hl, e1_dst, e1_src, E1_, aggP1);
        k_mean<<<gridFor(nPH, B), B, 0, stream>>>(aggP1, cntP1, NP_);
        k_zero<<<gridFor(nPH, B), B, 0, stream>>>(aggP2, nPH);
        k_scatter<<<gridFor((long long)E2_ * 32, B), B, 0, stream>>>(hd, e2_dst, e2_src, E2_, aggP2);
        k_mean<<<gridFor(nPH, B), B, 0, stream>>>(aggP2, cntP2, NP_);
        k_gemm4<<<gridFor(NP_, 16), 256, 0, stream>>>(
            aggP1, Wl + 2 * HH_, aggP2, Wl + 3 * HH_, hp, Wr + 2 * HH_, hp, Wr + 3 * HH_,
            bl + 2 * H_, bl + 3 * H_, hp2, NP_, H_, 0);

        // BN + ReLU (patients, labs, diseases)
        struct { const float* x; float* y; int N; const float* g; const float* b; } bn[3] = {
            { hp2, hp, NP_, g + 0 * H_, bb + 0 * H_ },
            { hl2, hl, NL_, g + 1 * H_, bb + 1 * H_ },
            { hd2, hd, ND_, g + 2 * H_, bb + 2 * H_ },
        };
        for (int i = 0; i < 3; ++i) {
            k_zero<<<1, H_, 0, stream>>>(bn_sum, H_);
            k_zero<<<1, H_, 0, stream>>>(bn_sq, H_);
            int sblocks = bn[i].N < 2048 ? bn[i].N : 2048;
            k_bn_stats<<<sblocks, H_, 0, stream>>>(bn[i].x, bn[i].N, bn_sum, bn_sq);
            k_bn_prep<<<1, H_, 0, stream>>>(bn_sum, bn_sq, bn[i].g, bn[i].b,
                                            1.0f / (float)bn[i].N, bn_scale, bn_shift);
            long long n = (long long)bn[i].N * H_;
            k_bn_apply_relu<<<gridFor(n, B), B, 0, stream>>>(bn[i].x, bn_scale, bn_shift, bn[i].y, n);
        }
    }

    // ---- head: relu(hp @ fc1_W + fc1_b) @ fc2_W + fc2_b -> log_softmax ----
    k_gemm4<<<gridFor(NP_, 16), 256, 0, stream>>>(
        hp, fc1_W, nullptr, nullptr, nullptr, nullptr, nullptr, nullptr,
        fc1_b, nullptr, fc1out, NP_, FC1_, 1);
    k_head<<<gridFor(NP_, B), B, 0, stream>>>(fc1out, fc2_W, fc2_b, (float*)d_out);
}